// LightGCN_82489141887090
// MI455X (gfx1250) — compile-verified
//
#include <hip/hip_runtime.h>
#include <hip/hip_bf16.h>

#define NUSERS      500000
#define AUDIO_SCALE 1.0f
#define META_SCALE  0.5f
#define EPSF        1e-12f

typedef __attribute__((ext_vector_type(2))) float v2f;
typedef __attribute__((ext_vector_type(8))) float v8f;

// ---------------------------------------------------------------- utilities
__global__ void k_fill(float* __restrict__ p, float v, long long n) {
    long long g = (long long)blockIdx.x * blockDim.x + threadIdx.x;
    if (g < n) p[g] = v;
}

// mark nodes that already have a self-loop: loop weight becomes 0
__global__ void k_selfloop(const int* __restrict__ row, const int* __restrict__ col,
                           float* __restrict__ selfw, int E) {
    int e = blockIdx.x * blockDim.x + threadIdx.x;
    if (e >= E) return;
    int r = row[e];
    if (r == col[e]) selfw[r] = 0.0f;   // all writers write 0: race-free
}

// deg[col] += w  (existing edges only; appended self-loops handled in k_dinv)
__global__ void k_deg(const int* __restrict__ col, const float* __restrict__ w,
                      float* __restrict__ deg, int E) {
    int e = blockIdx.x * blockDim.x + threadIdx.x;
    if (e >= E) return;
    atomicAdd(&deg[col[e]], w[e]);
}

__global__ void k_dinv(const float* __restrict__ deg, const float* __restrict__ selfw,
                       float* __restrict__ dinv, float* __restrict__ snorm, int N) {
    int i = blockIdx.x * blockDim.x + threadIdx.x;
    if (i >= N) return;
    float d  = deg[i] + selfw[i];                     // total degree incl. appended loop
    float di = (d > 0.0f) ? rsqrtf(fmaxf(d, EPSF)) : 0.0f;
    dinv[i]  = di;
    snorm[i] = di * di * selfw[i];                    // norm of the appended self-loop
}

__global__ void k_norm(const int* __restrict__ row, const int* __restrict__ col,
                       const float* __restrict__ w, const float* __restrict__ dinv,
                       float* __restrict__ norm, int E) {
    int e = blockIdx.x * blockDim.x + threadIdx.x;
    if (e >= E) return;
    norm[e] = dinv[row[e]] * w[e] * dinv[col[e]];
}

// ------------------------------------------------------- feature construction
// Block = 128 threads = 4 waves; each wave builds a 16x128 A-tile in LDS and
// computes a 16x64 projection tile with v_wmma_f32_16x16x4_f32 (32 K-steps x
// 4 N-tiles), then nodes are finalized (user vs item select + L2 norm).
__global__ __launch_bounds__(128) void k_features(
    const int* __restrict__ batch_nodes,
    const float* __restrict__ user_emb,
    const float* __restrict__ artist_emb,
    const float* __restrict__ album_emb,
    const float* __restrict__ audio_emb,
    const int* __restrict__ artist_ids,
    const int* __restrict__ album_ids,
    const float* __restrict__ proj_w,    // [128][64] row-major
    const float* __restrict__ proj_b,    // [64]
    float* __restrict__ x0,
    float* __restrict__ acc,
    int N)
{
    __shared__ float A[4][16][132];      // 16x128 per wave, padded
    __shared__ float Dt[4][16][65];      // 16x64 per wave, padded

    const int tid  = threadIdx.x;
    const int w    = tid >> 5;           // wave in block
    const int l    = tid & 31;           // lane (wave32)
    const int m    = l & 15;             // row / column index inside tile
    const int half = l >> 4;             // lane-half selects K pairs / M halves

    // ---- stage A-tile: lane handles row m; half 0 = audio cols, half 1 = meta
    {
        int grow = blockIdx.x * 64 + w * 16 + m;
        if (grow >= N) grow = N - 1;     // clamp: keep EXEC all-ones for WMMA
        int node = batch_nodes[grow];
        int i    = (node < NUSERS) ? 0 : (node - NUSERS);
        if (half == 0) {
            const float* src = audio_emb + (size_t)i * 64;
            #pragma unroll 8
            for (int d = 0; d < 64; ++d) A[w][m][d] = src[d] * AUDIO_SCALE;
        } else {
            const float* s0 = artist_emb + (size_t)artist_ids[i] * 64;
            const float* s1 = album_emb  + (size_t)album_ids[i]  * 64;
            #pragma unroll 8
            for (int d = 0; d < 64; ++d) A[w][m][64 + d] = (s0[d] + s1[d]) * META_SCALE;
        }
    }
    __syncthreads();

    // ---- 16x64 = A(16x128) * W(128x64) via V_WMMA_F32_16X16X4_F32
    // A frag (16x4 f32, 2 VGPRs): lanes 0-15 hold K={4k,4k+1}, lanes 16-31 K={4k+2,4k+3}
    // B frag (4x16 f32, 2 VGPRs): lane n holds B[k][n] with same K split
    v8f c0 = {}, c1 = {}, c2 = {}, c3 = {};
    #pragma unroll 4
    for (int kk = 0; kk < 32; ++kk) {
        const int kb = kk * 4 + half * 2;
        v2f a;
        a.x = A[w][m][kb];
        a.y = A[w][m][kb + 1];
        const float* wr0 = proj_w + (size_t)kb * 64 + m;
        const float* wr1 = wr0 + 64;
        v2f b0, b1, b2, b3;
        b0.x = wr0[0];  b0.y = wr1[0];
        b1.x = wr0[16]; b1.y = wr1[16];
        b2.x = wr0[32]; b2.y = wr1[32];
        b3.x = wr0[48]; b3.y = wr1[48];
        c0 = __builtin_amdgcn_wmma_f32_16x16x4_f32(false, a, false, b0, (short)0, c0, false, false);
        c1 = __builtin_amdgcn_wmma_f32_16x16x4_f32(false, a, false, b1, (short)0, c1, false, false);
        c2 = __builtin_amdgcn_wmma_f32_16x16x4_f32(false, a, false, b2, (short)0, c2, false, false);
        c3 = __builtin_amdgcn_wmma_f32_16x16x4_f32(false, a, false, b3, (short)0, c3, false, false);
    }

    // ---- spill D tile (+bias) to LDS: VGPR r -> M = r + half*8, N = lane&15
    {
        const int mo = half * 8;
        #pragma unroll
        for (int r = 0; r < 8; ++r) {
            Dt[w][mo + r][m +  0] = c0[r] + proj_b[m +  0];
            Dt[w][mo + r][m + 16] = c1[r] + proj_b[m + 16];
            Dt[w][mo + r][m + 32] = c2[r] + proj_b[m + 32];
            Dt[w][mo + r][m + 48] = c3[r] + proj_b[m + 48];
        }
    }
    __syncthreads();

    // ---- finalize: threads 0..63 each own one node (select user/item, L2 norm)
    if (tid < 64) {
        int gnode = blockIdx.x * 64 + tid;
        if (gnode < N) {
            int node = batch_nodes[gnode];
            float v[64];
            if (node < NUSERS) {
                const float* src = user_emb + (size_t)node * 64;
                #pragma unroll 8
                for (int d = 0; d < 64; ++d) v[d] = src[d];
            } else {
                const int ww = tid >> 4, rr = tid & 15;
                #pragma unroll 8
                for (int d = 0; d < 64; ++d) v[d] = Dt[ww][rr][d];
            }
            float s = 0.0f;
            #pragma unroll 8
            for (int d = 0; d < 64; ++d) s += v[d] * v[d];
            float inv = 1.0f / fmaxf(sqrtf(s), EPSF);
            float* o0 = x0  + (size_t)gnode * 64;
            float* o1 = acc + (size_t)gnode * 64;
            #pragma unroll 8
            for (int d = 0; d < 64; ++d) { float t = v[d] * inv; o0[d] = t; o1[d] = t; }
        }
    }
}

// --------------------------------------------------------------- propagation
// one thread per (edge, dim): y[col] += norm * x[row]
__global__ void k_scatter(const int* __restrict__ row, const int* __restrict__ col,
                          const float* __restrict__ norm, const float* __restrict__ x,
                          float* __restrict__ y, int E) {
    size_t g = (size_t)blockIdx.x * blockDim.x + threadIdx.x;
    if (g >= (size_t)E * 64) return;
    int e = (int)(g >> 6);
    int d = (int)(g & 63);
    float val = norm[e] * x[(size_t)row[e] * 64 + d];
    atomicAdd(&y[(size_t)col[e] * 64 + d], val);
}

// x1 += snorm * x0 (appended self-loops, no atomics needed); acc += x1
__global__ void k_selfacc(const float* __restrict__ snorm, const float* __restrict__ x0,
                          float* __restrict__ x1, float* __restrict__ acc, int N) {
    size_t g = (size_t)blockIdx.x * blockDim.x + threadIdx.x;
    if (g >= (size_t)N * 64) return;
    int i = (int)(g >> 6);
    float t = x1[g] + snorm[i] * x0[g];
    x1[g] = t;
    acc[g] += t;
}

// in-place: out = l2norm(acc / 4)
__global__ void k_final(float* __restrict__ out, int N) {
    int i = blockIdx.x * blockDim.x + threadIdx.x;
    if (i >= N) return;
    float* p = out + (size_t)i * 64;
    float v[64];
    float s = 0.0f;
    #pragma unroll 8
    for (int d = 0; d < 64; ++d) { v[d] = p[d] * 0.25f; s += v[d] * v[d]; }
    float inv = 1.0f / fmaxf(sqrtf(s), EPSF);
    #pragma unroll 8
    for (int d = 0; d < 64; ++d) p[d] = v[d] * inv;
}

// ------------------------------------------------------------------- launch
extern "C" void kernel_launch(void* const* d_in, const int* in_sizes, int n_in,
                              void* d_out, int out_size, void* d_ws, size_t ws_size,
                              hipStream_t stream) {
    const int*   batch_nodes = (const int*)d_in[0];
    const int*   edge_index  = (const int*)d_in[1];
    const float* edge_w      = (const float*)d_in[2];
    const float* user_emb    = (const float*)d_in[3];
    const float* artist_emb  = (const float*)d_in[4];
    const float* album_emb   = (const float*)d_in[5];
    const float* audio_emb   = (const float*)d_in[6];
    const int*   artist_ids  = (const int*)d_in[7];
    const int*   album_ids   = (const int*)d_in[8];
    const float* proj_w      = (const float*)d_in[9];
    const float* proj_b      = (const float*)d_in[10];

    const int N = in_sizes[0];
    const int E = in_sizes[2];
    const int* row = edge_index;
    const int* col = edge_index + E;

    char* ws = (char*)d_ws;
    size_t off = 0;
    auto carve = [&](size_t bytes) -> void* {
        void* p = ws + off;
        off += (bytes + 255) & ~(size_t)255;
        return p;
    };
    float* selfw = (float*)carve((size_t)N * 4);
    float* deg   = (float*)carve((size_t)N * 4);
    float* dinv  = (float*)carve((size_t)N * 4);
    float* snorm = (float*)carve((size_t)N * 4);
    float* norm  = (float*)carve((size_t)E * 4);
    float* x0    = (float*)carve((size_t)N * 64 * 4);
    float* x1    = (float*)carve((size_t)N * 64 * 4);
    float* acc   = (float*)d_out;   // accumulator lives in the output buffer

    const int T = 256;
    const int bN  = (N + T - 1) / T;
    const int bE  = (E + T - 1) / T;
    const long long nF = (long long)N * 64;
    const int bNF = (int)((nF + T - 1) / T);
    const long long eF = (long long)E * 64;
    const int bEF = (int)((eF + T - 1) / T);

    // normalization setup
    k_fill<<<bN, T, 0, stream>>>(selfw, 1.0f, (long long)N);
    k_fill<<<bN, T, 0, stream>>>(deg, 0.0f, (long long)N);
    k_selfloop<<<bE, T, 0, stream>>>(row, col, selfw, E);
    k_deg<<<bE, T, 0, stream>>>(col, edge_w, deg, E);
    k_dinv<<<bN, T, 0, stream>>>(deg, selfw, dinv, snorm, N);
    k_norm<<<bE, T, 0, stream>>>(row, col, edge_w, dinv, norm, E);

    // layer-0 features (WMMA item projection), also seeds the accumulator
    k_features<<<(N + 63) / 64, 128, 0, stream>>>(
        batch_nodes, user_emb, artist_emb, album_emb, audio_emb,
        artist_ids, album_ids, proj_w, proj_b, x0, acc, N);

    // 3 LightGCN layers
    float* xa = x0;
    float* xb = x1;
    for (int layer = 0; layer < 3; ++layer) {
        k_fill<<<bNF, T, 0, stream>>>(xb, 0.0f, nF);
        k_scatter<<<bEF, T, 0, stream>>>(row, col, norm, xa, xb, E);
        k_selfacc<<<bNF, T, 0, stream>>>(snorm, xa, xb, acc, N);
        float* tmp = xa; xa = xb; xb = tmp;
    }

    // out = l2norm(acc / 4), in place
    k_final<<<bN, T, 0, stream>>>(acc, N);
}